// Linear_Attention_82695300317577
// MI455X (gfx1250) — compile-verified
//
#include <hip/hip_runtime.h>
#include <stddef.h>
#include <stdint.h>

// ---------------------------------------------------------------------------
// Linear attention, fp32, CDNA5 (gfx1250):
//   - V_WMMA_F32_16X16X4_F32 for all matrix math (fp32-exact)
//   - Tensor Data Mover (tensor_load_to_lds) + TENSORcnt double buffering
//     for GEMM tile staging, with hardware LDS padding for conflict-free
//     WMMA fragment reads.
// ---------------------------------------------------------------------------

typedef __attribute__((ext_vector_type(2))) float v2f;
typedef __attribute__((ext_vector_type(8))) float v8f;
typedef __attribute__((ext_vector_type(4))) unsigned int u32x4;
typedef __attribute__((ext_vector_type(8))) int i32x8;
typedef __attribute__((ext_vector_type(4))) int i32x4;

#define DIMX   1024
#define HEADS  16
#define DHEAD  64
#define BATCH  4
#define SEQ    4096
#define ROWS   (BATCH * SEQ)   // 16384
#define NQKV   (3 * DIMX)      // 3072
#define KV_CHUNK  512
#define KV_NCHUNK (SEQ / KV_CHUNK)  // 8

#if __has_builtin(__builtin_amdgcn_tensor_load_to_lds) && \
    __has_builtin(__builtin_amdgcn_s_wait_tensorcnt)
#define USE_TDM 1
#else
#define USE_TDM 0
#endif

__device__ __forceinline__ v8f wmma4(v2f a, v2f b, v8f c) {
  // D = A(16x4 f32) * B(4x16 f32) + C(16x16 f32), wave32
  return __builtin_amdgcn_wmma_f32_16x16x4_f32(false, a, false, b, (short)0, c,
                                               false, false);
}

#if USE_TDM
// Generic (flat) pointers to LDS carry the workgroup-relative LDS offset in
// their low 32 bits on AMDGPU.
__device__ __forceinline__ unsigned lds_offset_of(const void* p) {
  return (unsigned)(uintptr_t)p;
}

// Issue one TDM 2-D tile load: tile_x elements (f32, contiguous) per row,
// tile_y rows, row stride `stride_elems` in global memory; destination LDS
// gets `pad_amt` dwords inserted after every `pad_int` dwords
// (codes per D# group1: interval 0=2,1=4,2=8,3=16,4=32,5=64,6=128,7=256
//  dwords; amount 0=1 .. 127=128 dwords).
__device__ __forceinline__ void tdm_load_2d_f32(
    unsigned lds_byte_off, const float* gptr, unsigned tile_x, unsigned tile_y,
    unsigned stride_elems, unsigned pad_int_code, unsigned pad_amt_code) {
  const unsigned long long ga = (unsigned long long)(uintptr_t)gptr;
  u32x4 g0;
  g0[0] = 1u;                                   // count=1 (valid descriptor)
  g0[1] = lds_byte_off;                         // lds_addr  (bits 63:32)
  g0[2] = (unsigned)ga;                         // global_addr[31:0]  (95:64)
  g0[3] = ((unsigned)(ga >> 32) & 0x1FFFFFFu)   // global_addr[56:32] (120:96)
          | (2u << 30);                         // type=2 ("image")   (127:126)
  i32x8 g1;
  g1[0] = (int)((2u << 16)                      // data_size = 4 bytes
                | (1u << 20)                    // pad_enable
                | (pad_int_code << 22)          // pad_interval
                | (pad_amt_code << 25));        // pad_amount
  g1[1] = (int)((tile_x & 0xFFFFu) << 16);      // tensor_dim0[15:0] @ 63:48
  g1[2] = (int)(((tile_x >> 16) & 0xFFFFu)      // tensor_dim0[31:16]
                | ((tile_y & 0xFFFFu) << 16));  // tensor_dim1[15:0]
  g1[3] = (int)(((tile_y >> 16) & 0xFFFFu)      // tensor_dim1[31:16]
                | ((tile_x & 0xFFFFu) << 16));  // tile_dim0 @ 127:112
  g1[4] = (int)(tile_y & 0xFFFFu);              // tile_dim1; tile_dim2=0
  g1[5] = (int)stride_elems;                    // tensor_dim0_stride[31:0]
  g1[6] = 0;                                    // stride[47:32]=0, dim1_stride lo
  g1[7] = 0;
  const i32x4 z4 = {0, 0, 0, 0};                // 2-D: groups 2/3 unused
  const i32x8 z8 = {0, 0, 0, 0, 0, 0, 0, 0};    // extension group unused
  __builtin_amdgcn_tensor_load_to_lds(g0, g1, z4, z4, z8, 0);
}
#endif  // USE_TDM

// ---------------------------------------------------------------------------
// Kernel: generic fp32 WMMA GEMM  C[M,N] = A[M,K] @ B[K,N] (+ bias)
// Block tile 128x128, BK=16, 8 waves (256 thr), wave tile 32x64 (2x4 wmma).
// TDM double-buffered tile staging; fragment reads are conflict-free thanks
// to the hardware-padded LDS strides (A: 16+2=18, B: 128+8=136 dwords).
// M,N multiples of 128; K multiple of 16 (true for all uses here).
// ---------------------------------------------------------------------------
template <bool HAS_BIAS>
__global__ __launch_bounds__(256) void gemm_f32_wmma(
    const float* __restrict__ A, const float* __restrict__ B,
    const float* __restrict__ bias, float* __restrict__ C,
    int M, int N, int K) {
  constexpr int BM = 128, BN = 128, BK = 16;
  constexpr int AS = 18;   // LDS word-stride of A tile (bank-conflict-free b64)
  constexpr int BS = 136;  // LDS word-stride of B tile (halves -> disjoint banks)

  const int bm = blockIdx.y * BM;
  const int bn = blockIdx.x * BN;
  const int tid = threadIdx.x;
  const int lane = tid & 31;
  const int wave = tid >> 5;
  const int lm = lane & 15;     // fragment lane index (M or N)
  const int half = lane >> 4;   // lane-half selects K pair
  const int wm = (wave & 3) * 32;   // wave M origin in tile
  const int wn = (wave >> 2) * 64;  // wave N origin in tile

  v8f acc[2][4];
  const v8f vzero = {0.f, 0.f, 0.f, 0.f, 0.f, 0.f, 0.f, 0.f};
#pragma unroll
  for (int i = 0; i < 2; ++i)
#pragma unroll
    for (int j = 0; j < 4; ++j) acc[i][j] = vzero;

#if USE_TDM
  __shared__ float sA[2][BM * AS];
  __shared__ float sB[2][BK * BS];

  // Prologue: wave 0 launches the DMA for the first K tile.
  if (wave == 0) {
    tdm_load_2d_f32(lds_offset_of(&sA[0][0]), A + (size_t)bm * K,
                    BK, BM, (unsigned)K, /*16dw*/ 3, /*+2dw*/ 1);
    tdm_load_2d_f32(lds_offset_of(&sB[0][0]), B + bn,
                    BN, BK, (unsigned)N, /*128dw*/ 6, /*+8dw*/ 7);
  }

  int buf = 0;
  for (int k0 = 0; k0 < K; k0 += BK) {
    if (wave == 0) {
      if (k0 + BK < K) {
        // Kick off the next tile's DMA, then wait only for the current one
        // (tensor ops from one wave complete in order => wait count <= 2).
        tdm_load_2d_f32(lds_offset_of(&sA[buf ^ 1][0]),
                        A + (size_t)bm * K + (k0 + BK), BK, BM, (unsigned)K,
                        3, 1);
        tdm_load_2d_f32(lds_offset_of(&sB[buf ^ 1][0]),
                        B + (size_t)(k0 + BK) * N + bn, BN, BK, (unsigned)N,
                        6, 7);
        __builtin_amdgcn_s_wait_tensorcnt(2);
      } else {
        __builtin_amdgcn_s_wait_tensorcnt(0);
      }
    }
    __syncthreads();  // current buffer resident for all waves

    const float* __restrict__ tA = sA[buf];
    const float* __restrict__ tB = sB[buf];
#pragma unroll
    for (int kk = 0; kk < BK; kk += 4) {
      v2f af[2];
      v2f bf[4];
#pragma unroll
      for (int tm = 0; tm < 2; ++tm) {
        const float* p = &tA[(wm + tm * 16 + lm) * AS + kk + 2 * half];
        af[tm] = (v2f){p[0], p[1]};  // contiguous b64 per lane
      }
#pragma unroll
      for (int tn = 0; tn < 4; ++tn) {
        const int n = wn + tn * 16 + lm;
        bf[tn] = (v2f){tB[(kk + 2 * half) * BS + n],
                       tB[(kk + 2 * half + 1) * BS + n]};
      }
#pragma unroll
      for (int tm = 0; tm < 2; ++tm)
#pragma unroll
        for (int tn = 0; tn < 4; ++tn)
          acc[tm][tn] = wmma4(af[tm], bf[tn], acc[tm][tn]);
    }
    __syncthreads();  // done with `buf`; DMA may overwrite it next iteration
    buf ^= 1;
  }
#else  // !USE_TDM: synchronous staging fallback
  __shared__ float sA[BM * AS];
  __shared__ float sB[BK * BS];
  for (int k0 = 0; k0 < K; k0 += BK) {
#pragma unroll
    for (int p = 0; p < 2; ++p) {
      const int idx = tid + p * 256;
      const int r = idx >> 2;
      const int c = (idx & 3) << 2;
      const float4 v = *(const float4*)(A + (size_t)(bm + r) * K + k0 + c);
      sA[r * AS + c + 0] = v.x;
      sA[r * AS + c + 1] = v.y;
      sA[r * AS + c + 2] = v.z;
      sA[r * AS + c + 3] = v.w;
    }
#pragma unroll
    for (int p = 0; p < 2; ++p) {
      const int idx = tid + p * 256;
      const int r = idx >> 5;
      const int c = (idx & 31) << 2;
      const float4 v = *(const float4*)(B + (size_t)(k0 + r) * N + bn + c);
      sB[r * BS + c + 0] = v.x;
      sB[r * BS + c + 1] = v.y;
      sB[r * BS + c + 2] = v.z;
      sB[r * BS + c + 3] = v.w;
    }
    __syncthreads();
#pragma unroll
    for (int kk = 0; kk < BK; kk += 4) {
      v2f af[2];
      v2f bf[4];
#pragma unroll
      for (int tm = 0; tm < 2; ++tm) {
        const float* p = &sA[(wm + tm * 16 + lm) * AS + kk + 2 * half];
        af[tm] = (v2f){p[0], p[1]};
      }
#pragma unroll
      for (int tn = 0; tn < 4; ++tn) {
        const int n = wn + tn * 16 + lm;
        bf[tn] = (v2f){sB[(kk + 2 * half) * BS + n],
                       sB[(kk + 2 * half + 1) * BS + n]};
      }
#pragma unroll
      for (int tm = 0; tm < 2; ++tm)
#pragma unroll
        for (int tn = 0; tn < 4; ++tn)
          acc[tm][tn] = wmma4(af[tm], bf[tn], acc[tm][tn]);
    }
    __syncthreads();
  }
#endif  // USE_TDM

  // ---- store (C/D layout: VGPR r + 8*half -> row, lane&15 -> col) ----
#pragma unroll
  for (int tm = 0; tm < 2; ++tm) {
#pragma unroll
    for (int tn = 0; tn < 4; ++tn) {
      const int col = bn + wn + tn * 16 + lm;
      const float bv = HAS_BIAS ? bias[col] : 0.f;
#pragma unroll
      for (int r = 0; r < 8; ++r) {
        const int row = bm + wm + tm * 16 + r + 8 * half;
        C[(size_t)row * N + col] = acc[tm][tn][r] + bv;
      }
    }
  }
}

// ---------------------------------------------------------------------------
// Kernel: in-place rmsnorm (+weight) + ReLU over q and k segments of qkv rows.
// One block (256 thr) per row; each segment is 1024 floats (4 per thread).
// ---------------------------------------------------------------------------
__global__ __launch_bounds__(256) void rmsnorm_relu_kernel(
    float* __restrict__ qkv, const float* __restrict__ qn_w,
    const float* __restrict__ kn_w) {
  const int row = blockIdx.x;
  const int tid = threadIdx.x;
  __shared__ float red[8];

#pragma unroll
  for (int seg = 0; seg < 2; ++seg) {
    const float* w = (seg == 0) ? qn_w : kn_w;
    float* p = qkv + (size_t)row * NQKV + seg * DIMX;
    const float4 v = *(const float4*)(p + tid * 4);
    float ss = v.x * v.x + v.y * v.y + v.z * v.z + v.w * v.w;
#pragma unroll
    for (int off = 16; off > 0; off >>= 1) ss += __shfl_down(ss, off, 32);
    if ((tid & 31) == 0) red[tid >> 5] = ss;
    __syncthreads();
    float tot = 0.f;
#pragma unroll
    for (int i = 0; i < 8; ++i) tot += red[i];
    const float rs = rsqrtf(tot * (1.0f / (float)DIMX) + 1e-6f);
    const float4 wv = *(const float4*)(w + tid * 4);
    float4 o;
    o.x = fmaxf(v.x * rs * wv.x, 0.f);
    o.y = fmaxf(v.y * rs * wv.y, 0.f);
    o.z = fmaxf(v.z * rs * wv.z, 0.f);
    o.w = fmaxf(v.w * rs * wv.w, 0.f);
    *(float4*)(p + tid * 4) = o;
    __syncthreads();  // red[] reused by next segment
  }
}

// ---------------------------------------------------------------------------
// Kernel: per (b,h,chunk): partial kv = K_chunk^T @ V_chunk (64x64) + col sums.
// grid.x = BATCH*HEADS*KV_NCHUNK. 8 waves; wave tile 16x32 (2 wmma tiles).
// A-fragment is K transposed, read column-wise out of LDS.
// ---------------------------------------------------------------------------
__global__ __launch_bounds__(256) void kv_ksum_kernel(
    const float* __restrict__ qkv, float* __restrict__ kvpart,
    float* __restrict__ kspart) {
  constexpr int NT = 32;  // n rows staged per step
  constexpr int TS = 80;  // LDS word-stride (halves -> disjoint banks)
  __shared__ float sK[NT * TS];
  __shared__ float sV[NT * TS];

  const int bh = blockIdx.x / KV_NCHUNK;
  const int chunk = blockIdx.x % KV_NCHUNK;
  const int b = bh / HEADS, h = bh % HEADS;
  const int n0 = chunk * KV_CHUNK;
  const float* kbase = qkv + (size_t)b * SEQ * NQKV + DIMX + h * DHEAD;
  const float* vbase = qkv + (size_t)b * SEQ * NQKV + 2 * DIMX + h * DHEAD;

  const int tid = threadIdx.x;
  const int lane = tid & 31, wave = tid >> 5;
  const int lm = lane & 15, half = lane >> 4;
  const int wm = (wave & 3) * 16;   // d origin (rows of kv)
  const int wn = (wave >> 2) * 32;  // e origin (cols of kv)

  const v8f vzero = {0.f, 0.f, 0.f, 0.f, 0.f, 0.f, 0.f, 0.f};
  v8f acc[2] = {vzero, vzero};
  float ks = 0.f;  // per-thread (tid<64) running column sum of K

  for (int nn = 0; nn < KV_CHUNK; nn += NT) {
    // stage 32x64 K and V tiles (512 float4 each, 2 per thread)
#pragma unroll
    for (int p = 0; p < 2; ++p) {
      const int idx = tid + p * 256;
      const int r = idx >> 4;
      const int c = (idx & 15) << 2;
      const size_t go = (size_t)(n0 + nn + r) * NQKV + c;
      const float4 a = *(const float4*)(kbase + go);
      const float4 v = *(const float4*)(vbase + go);
      sK[r * TS + c + 0] = a.x; sK[r * TS + c + 1] = a.y;
      sK[r * TS + c + 2] = a.z; sK[r * TS + c + 3] = a.w;
      sV[r * TS + c + 0] = v.x; sV[r * TS + c + 1] = v.y;
      sV[r * TS + c + 2] = v.z; sV[r * TS + c + 3] = v.w;
    }
    __syncthreads();

    if (tid < DHEAD) {
#pragma unroll
      for (int i = 0; i < NT; ++i) ks += sK[i * TS + tid];
    }

#pragma unroll
    for (int kk = 0; kk < NT; kk += 4) {
      // A[d][n] = K[n][d] : transpose out of LDS
      const v2f a = (v2f){sK[(kk + 2 * half) * TS + wm + lm],
                          sK[(kk + 2 * half + 1) * TS + wm + lm]};
      const v2f b0 = (v2f){sV[(kk + 2 * half) * TS + wn + lm],
                           sV[(kk + 2 * half + 1) * TS + wn + lm]};
      const v2f b1 = (v2f){sV[(kk + 2 * half) * TS + wn + 16 + lm],
                           sV[(kk + 2 * half + 1) * TS + wn + 16 + lm]};
      acc[0] = wmma4(a, b0, acc[0]);
      acc[1] = wmma4(a, b1, acc[1]);
    }
    __syncthreads();
  }

  // write partials (deterministic fixed-order reduction happens later)
  float* kvp = kvpart + (size_t)blockIdx.x * DHEAD * DHEAD;
#pragma unroll
  for (int tn = 0; tn < 2; ++tn) {
    const int e = wn + tn * 16 + lm;
#pragma unroll
    for (int r = 0; r < 8; ++r) {
      const int d = wm + r + 8 * half;
      kvp[d * DHEAD + e] = acc[tn][r];
    }
  }
  if (tid < DHEAD) kspart[(size_t)blockIdx.x * DHEAD + tid] = ks;
}

// ---------------------------------------------------------------------------
// Kernel: fixed-order reduction of kv / ksum partials over KV_NCHUNK chunks.
// ---------------------------------------------------------------------------
__global__ __launch_bounds__(256) void reduce_kv_kernel(
    const float* __restrict__ kvpart, const float* __restrict__ kspart,
    float* __restrict__ kv, float* __restrict__ ksum) {
  const int idx = blockIdx.x * 256 + threadIdx.x;
  if (idx < BATCH * HEADS * DHEAD * DHEAD) {
    const int bh = idx >> 12;  // / 4096
    const int i = idx & 4095;
    float s = 0.f;
#pragma unroll
    for (int c = 0; c < KV_NCHUNK; ++c)
      s += kvpart[((size_t)bh * KV_NCHUNK + c) * DHEAD * DHEAD + i];
    kv[idx] = s;
  }
  if (idx < BATCH * HEADS * DHEAD) {
    const int bh = idx >> 6;
    const int i = idx & 63;
    float s = 0.f;
#pragma unroll
    for (int c = 0; c < KV_NCHUNK; ++c)
      s += kspart[((size_t)bh * KV_NCHUNK + c) * DHEAD + i];
    ksum[idx] = s;
  }
}

// ---------------------------------------------------------------------------
// Kernel: per (b,h,128-row block): out = (Q @ kv) / (Q·k_sum + eps), written
// into attn[b,n,1024] (head-concat layout) for the final GEMM.
// 8 waves; wave = 16 rows x 64 cols (4 wmma tiles), K-dim = 64.
// ---------------------------------------------------------------------------
__global__ __launch_bounds__(256) void attn_out_kernel(
    const float* __restrict__ qkv, const float* __restrict__ kv,
    const float* __restrict__ ksum, float* __restrict__ attn) {
  constexpr int QS = 68;   // LDS stride for Q tile (b64 frag, conflict-free)
  constexpr int KVS = 80;  // LDS stride for kv tile
  __shared__ float sQ[128 * QS];
  __shared__ float sKV[DHEAD * KVS];
  __shared__ float sKS[DHEAD];
  __shared__ float sRN[128];

  const int nblks = SEQ / 128;
  const int bh = blockIdx.x / nblks;
  const int nb = blockIdx.x % nblks;
  const int b = bh / HEADS, h = bh % HEADS;
  const int row0 = nb * 128;

  const int tid = threadIdx.x;
  const int lane = tid & 31, wave = tid >> 5;
  const int lm = lane & 15, half = lane >> 4;

  const float* qbase = qkv + (size_t)b * SEQ * NQKV + h * DHEAD;  // q @ col 0
  const float* kvbh = kv + (size_t)bh * DHEAD * DHEAD;

  // stage kv (64x64 = 1024 float4, 4 per thread)
#pragma unroll
  for (int p = 0; p < 4; ++p) {
    const int idx = tid + p * 256;
    const int r = idx >> 4;
    const int c = (idx & 15) << 2;
    const float4 v = *(const float4*)(kvbh + r * DHEAD + c);
    sKV[r * KVS + c + 0] = v.x; sKV[r * KVS + c + 1] = v.y;
    sKV[r * KVS + c + 2] = v.z; sKV[r * KVS + c + 3] = v.w;
  }
  if (tid < DHEAD) sKS[tid] = ksum[(size_t)bh * DHEAD + tid];

  // stage Q tile (128x64 = 2048 float4, 8 per thread)
#pragma unroll
  for (int p = 0; p < 8; ++p) {
    const int idx = tid + p * 256;
    const int r = idx >> 4;
    const int c = (idx & 15) << 2;
    const float4 v = *(const float4*)(qbase + (size_t)(row0 + r) * NQKV + c);
    sQ[r * QS + c + 0] = v.x; sQ[r * QS + c + 1] = v.y;
    sQ[r * QS + c + 2] = v.z; sQ[r * QS + c + 3] = v.w;
  }
  __syncthreads();

  // reciprocal normalizer per row
  if (tid < 128) {
    float s = 0.f;
#pragma unroll
    for (int d = 0; d < DHEAD; ++d) s += sQ[tid * QS + d] * sKS[d];
    sRN[tid] = 1.0f / (s + 1e-6f);
  }
  __syncthreads();

  const v8f vzero = {0.f, 0.f, 0.f, 0.f, 0.f, 0.f, 0.f, 0.f};
  v8f acc[4] = {vzero, vzero, vzero, vzero};
#pragma unroll
  for (int kk = 0; kk < DHEAD; kk += 4) {
    const float* ap = &sQ[(wave * 16 + lm) * QS + kk + 2 * half];
    const v2f a = (v2f){ap[0], ap[1]};
#pragma unroll
    for (int tn = 0; tn < 4; ++tn) {
      const v2f bf = (v2f){sKV[(kk + 2 * half) * KVS + tn * 16 + lm],
                           sKV[(kk + 2 * half + 1) * KVS + tn * 16 + lm]};
      acc[tn] = wmma4(a, bf, acc[tn]);
    }
  }

  float* out = attn + (size_t)b * SEQ * DIMX + h * DHEAD;
#pragma unroll
  for (int tn = 0; tn < 4; ++tn) {
    const int e = tn * 16 + lm;
#pragma unroll
    for (int r = 0; r < 8; ++r) {
      const int m = wave * 16 + r + 8 * half;
      out[(size_t)(row0 + m) * DIMX + e] = acc[tn][r] * sRN[m];
    }
  }
}

// ---------------------------------------------------------------------------
// Launch: no atomics, stream-ordered dependencies, deterministic.
// ws layout (bytes):
//   qkv      : ROWS*NQKV*4            = 201326592
//   attn     : ROWS*DIMX*4            =  67108864
//   kvpart   : 64*KV_NCHUNK*4096*4    =   8388608
//   kspart   : 64*KV_NCHUNK*64*4      =    131072
//   kv       : 64*4096*4              =   1048576
//   ksum     : 64*64*4                =     16384
// total ~ 278 MB
// ---------------------------------------------------------------------------
extern "C" void kernel_launch(void* const* d_in, const int* in_sizes, int n_in,
                              void* d_out, int out_size, void* d_ws,
                              size_t ws_size, hipStream_t stream) {
  (void)in_sizes; (void)n_in; (void)out_size; (void)ws_size;
  const float* x     = (const float*)d_in[0];
  const float* Wqkv  = (const float*)d_in[1];
  const float* qn_w  = (const float*)d_in[2];
  const float* kn_w  = (const float*)d_in[3];
  const float* Wout  = (const float*)d_in[4];
  const float* bout  = (const float*)d_in[5];
  float* out = (float*)d_out;

  char* ws = (char*)d_ws;
  float* qkvb   = (float*)ws;
  size_t off = (size_t)ROWS * NQKV * 4;
  float* attn   = (float*)(ws + off);  off += (size_t)ROWS * DIMX * 4;
  float* kvpart = (float*)(ws + off);  off += (size_t)BATCH * HEADS * KV_NCHUNK * DHEAD * DHEAD * 4;
  float* kspart = (float*)(ws + off);  off += (size_t)BATCH * HEADS * KV_NCHUNK * DHEAD * 4;
  float* kvb    = (float*)(ws + off);  off += (size_t)BATCH * HEADS * DHEAD * DHEAD * 4;
  float* ksum   = (float*)(ws + off);

  // 1) qkv = x @ W_qkv
  {
    dim3 g(NQKV / 128, ROWS / 128);
    gemm_f32_wmma<false><<<g, 256, 0, stream>>>(x, Wqkv, nullptr, qkvb,
                                                ROWS, NQKV, DIMX);
  }
  // 2) rmsnorm + relu on q,k in place
  rmsnorm_relu_kernel<<<ROWS, 256, 0, stream>>>(qkvb, qn_w, kn_w);
  // 3) partial kv = K^T V and column sums, per (b,h,chunk)
  kv_ksum_kernel<<<BATCH * HEADS * KV_NCHUNK, 256, 0, stream>>>(qkvb, kvpart,
                                                                kspart);
  // 4) fixed-order reduce partials
  reduce_kv_kernel<<<(BATCH * HEADS * DHEAD * DHEAD + 255) / 256, 256, 0,
                     stream>>>(kvpart, kspart, kvb, ksum);
  // 5) out = (Q @ kv) / (Q . ksum + eps) into attn layout
  attn_out_kernel<<<BATCH * HEADS * (SEQ / 128), 256, 0, stream>>>(
      qkvb, kvb, ksum, attn);
  // 6) d_out = attn @ W_out + b_out
  {
    dim3 g(DIMX / 128, ROWS / 128);
    gemm_f32_wmma<true><<<g, 256, 0, stream>>>(attn, Wout, bout, out,
                                               ROWS, DIMX, DIMX);
  }
}